// SNetLoss_50861002719988
// MI455X (gfx1250) — compile-verified
//
#include <hip/hip_runtime.h>

#define D_DIM 160
#define H_DIM 192
#define W_DIM 160
#define NVOX (D_DIM * H_DIM * W_DIM)   // 4,915,200
#define HW   (H_DIM * W_DIM)
#define RAD 4
#define WIN_INV (1.0f / 729.0f)
#define EPS_LNCC 1e-5f

typedef float v2f __attribute__((ext_vector_type(2)));
typedef float v8f __attribute__((ext_vector_type(8)));

// ---------------- workspace layout -------------------------------------
// floats [0, 5*NVOX)                : 5 channels of WH-box-summed data
// doubles at byte 5*NVOX*4:
//   [0,    2400) cc per-block partials
//   [2400, 6240) gradD partials (3840)
//   [6240,10080) gradH partials (3840)
//   [10080,13920) gradW partials (3840)
#define NB_CC   2400
#define NB_GRAD 3840

__device__ __forceinline__ int iclamp(int v, int lo, int hi) {
    return v < lo ? lo : (v > hi ? hi : v);
}

// ---------------- Kernel A: pointwise products + W & H box sums --------
__global__ __launch_bounds__(256) void k_wh(const float* __restrict__ I,
                                            const float* __restrict__ J,
                                            float* __restrict__ out) {
    __shared__ float sI[24 * 24];
    __shared__ float sJ[24 * 24];
    __shared__ float sW[5][24 * 16];

    int bid = blockIdx.x;             // 19200 = 160 * 12 * 10
    int wT = bid % 10;
    int hT = (bid / 10) % 12;
    int d  = bid / 120;
    int h0 = hT * 16 - RAD;
    int w0 = wT * 16 - RAD;
    int tid = threadIdx.x;

    const float* Ib = I + (size_t)d * HW;
    const float* Jb = J + (size_t)d * HW;

    // Branchless halo load: clamp address, select zero for OOB.
    for (int t = tid; t < 24 * 24; t += 256) {
        int hh = t / 24, ww = t % 24;
        int gh = h0 + hh, gw = w0 + ww;
        bool ok = (gh >= 0) && (gh < H_DIM) && (gw >= 0) && (gw < W_DIM);
        unsigned off = (unsigned)(iclamp(gh, 0, H_DIM - 1) * W_DIM +
                                  iclamp(gw, 0, W_DIM - 1));
        float iv = Ib[off];
        float jv = Jb[off];
        sI[t] = ok ? iv : 0.0f;
        sJ[t] = ok ? jv : 0.0f;
    }
    __syncthreads();

    // W-direction 9-tap sums for 24 rows x 16 output cols, 5 channels
    for (int t = tid; t < 24 * 16; t += 256) {
        int hh = t / 16, wo = t % 16;
        int base = hh * 24 + wo;
        float aI = 0, aJ = 0, aII = 0, aJJ = 0, aIJ = 0;
        #pragma unroll
        for (int j = 0; j < 9; j++) {
            float iv = sI[base + j];
            float jv = sJ[base + j];
            aI += iv; aJ += jv;
            aII += iv * iv; aJJ += jv * jv; aIJ += iv * jv;
        }
        sW[0][t] = aI; sW[1][t] = aJ; sW[2][t] = aII; sW[3][t] = aJJ; sW[4][t] = aIJ;
    }
    __syncthreads();

    // H-direction 9-tap sums for the 16x16 output tile
    {
        int ho = tid / 16, wo = tid % 16;
        unsigned o = (unsigned)(((d * H_DIM) + hT * 16 + ho) * W_DIM + wT * 16 + wo);
        #pragma unroll
        for (int c = 0; c < 5; c++) {
            float a = 0.0f;
            #pragma unroll
            for (int j = 0; j < 9; j++) a += sW[c][(ho + j) * 16 + wo];
            out[(unsigned)c * (unsigned)NVOX + o] = a;   // 32-bit index, saddr form
        }
    }
}

// ---------------- Kernel B: D-axis box sum via WMMA + cc + reduce ------
// out(16d x 16w) tile = Band(16x24) x In(24d x 16w); six 16x16x4 f32 WMMAs
// per channel. Depth zero-padding is folded into the A band matrix so all
// B loads are unconditional (clamped, finite, multiplied by 0). All loads
// use a single 32-bit element index against one SGPR base (scale_offset).
__global__ __launch_bounds__(256) void k_d_cc(const float* __restrict__ S,
                                              double* __restrict__ partial) {
    __shared__ float red[8];
    int wave = threadIdx.x >> 5;
    int lane = threadIdx.x & 31;
    int tile = blockIdx.x * 8 + wave;   // 2400*8 = 19200 = 10 * 192 * 10
    int wT = tile % 10;
    int h  = (tile / 10) % H_DIM;
    int dT = tile / (10 * H_DIM);
    int dBase = dT * 16;
    int wBase = wT * 16;

    int m  = lane & 15;            // A row  (output d within tile)
    int kh = (lane >> 4) * 2;      // lanes 0-15 -> K+0,K+1 ; lanes 16-31 -> K+2,K+3
    int n  = lane & 15;            // B column (w within tile)
    unsigned colOff = (unsigned)(h * W_DIM + wBase + n);

    v8f acc5[5];
    #pragma unroll
    for (int c = 0; c < 5; c++) acc5[c] = (v8f){0.f,0.f,0.f,0.f,0.f,0.f,0.f,0.f};

    #pragma unroll
    for (int ch = 0; ch < 6; ch++) {
        int k0 = ch * 4 + kh;
        int k1 = k0 + 1;
        int g0 = dBase - RAD + k0;    // global input d row for K=k0
        int g1 = g0 + 1;
        bool in0 = (g0 >= 0) && (g0 < D_DIM);
        bool in1 = (g1 >= 0) && (g1 < D_DIM);
        v2f a;
        a.x = (in0 && k0 >= m && k0 <= m + 8) ? 1.0f : 0.0f;
        a.y = (in1 && k1 >= m && k1 <= m + 8) ? 1.0f : 0.0f;
        unsigned off0 = (unsigned)iclamp(g0, 0, D_DIM - 1) * (unsigned)HW + colOff;
        unsigned off1 = (unsigned)iclamp(g1, 0, D_DIM - 1) * (unsigned)HW + colOff;
        #pragma unroll
        for (int c = 0; c < 5; c++) {
            unsigned cb = (unsigned)c * (unsigned)NVOX;
            v2f b;
            b.x = S[cb + off0];       // 32-bit element index + SGPR base
            b.y = S[cb + off1];
            acc5[c] = __builtin_amdgcn_wmma_f32_16x16x4_f32(
                          false, a, false, b, (short)0, acc5[c], false, false);
        }
    }

    // cc per output element (8 rows of the tile live in this lane)
    float ccp = 0.0f;
    #pragma unroll
    for (int e = 0; e < 8; e++) {
        float Is = acc5[0][e], Js = acc5[1][e];
        float IIs = acc5[2][e], JJs = acc5[3][e], IJs = acc5[4][e];
        float cross = IJs - Is * Js * WIN_INV;
        float Iv = IIs - Is * Is * WIN_INV;
        float Jv = JJs - Js * Js * WIN_INV;
        ccp += cross * cross / (Iv * Jv + EPS_LNCC);
    }
    // deterministic wave reduce (wave32)
    #pragma unroll
    for (int off = 16; off >= 1; off >>= 1) ccp += __shfl_xor(ccp, off, 32);
    if (lane == 0) red[wave] = ccp;
    __syncthreads();
    if (threadIdx.x == 0) {
        double s = 0.0;
        #pragma unroll
        for (int i = 0; i < 8; i++) s += (double)red[i];
        partial[blockIdx.x] = s;
    }
}

// ---------------- Kernel C: gradient (finite-difference) loss ----------
// Each thread handles 4 consecutive w elements (float4 loads). OOB
// neighbors are redirected to self so the diff is exactly 0 (branchless).
__global__ __launch_bounds__(256) void k_grad(const float* __restrict__ s,
                                              double* __restrict__ partial) {
    float aD = 0.f, aH = 0.f, aW = 0.f;
    const int WQ = W_DIM / 4;                   // 40 groups per row
    const long totalq = 3L * NVOX / 4;
    for (long gid = blockIdx.x * 256L + threadIdx.x; gid < totalq;
         gid += (long)gridDim.x * 256L) {
        int wq = (int)(gid % WQ);
        long r = gid / WQ;
        int h = (int)(r % H_DIM);
        int d = (int)((r / H_DIM) % D_DIM);
        long idx = gid * 4;

        float4 self = *(const float4*)(s + idx);
        long offD = (d < D_DIM - 1) ? idx + HW : idx;
        long offH = (h < H_DIM - 1) ? idx + W_DIM : idx;
        float4 dn = *(const float4*)(s + offD);
        float4 rt = *(const float4*)(s + offH);
        float nxt = s[(wq < WQ - 1) ? idx + 4 : idx + 3];

        float t;
        t = dn.x - self.x; aD += t * t;
        t = dn.y - self.y; aD += t * t;
        t = dn.z - self.z; aD += t * t;
        t = dn.w - self.w; aD += t * t;
        t = rt.x - self.x; aH += t * t;
        t = rt.y - self.y; aH += t * t;
        t = rt.z - self.z; aH += t * t;
        t = rt.w - self.w; aH += t * t;
        t = self.y - self.x; aW += t * t;
        t = self.z - self.y; aW += t * t;
        t = self.w - self.z; aW += t * t;
        t = nxt - self.w;    aW += t * t;
    }
    __shared__ float rD[256], rH[256], rW[256];
    rD[threadIdx.x] = aD; rH[threadIdx.x] = aH; rW[threadIdx.x] = aW;
    __syncthreads();
    for (int off = 128; off > 0; off >>= 1) {
        if (threadIdx.x < off) {
            rD[threadIdx.x] += rD[threadIdx.x + off];
            rH[threadIdx.x] += rH[threadIdx.x + off];
            rW[threadIdx.x] += rW[threadIdx.x + off];
        }
        __syncthreads();
    }
    if (threadIdx.x == 0) {
        partial[NB_CC + blockIdx.x]               = (double)rD[0];
        partial[NB_CC + NB_GRAD + blockIdx.x]     = (double)rH[0];
        partial[NB_CC + 2 * NB_GRAD + blockIdx.x] = (double)rW[0];
    }
}

// ---------------- Kernel D: deterministic final combine ----------------
__global__ __launch_bounds__(256) void k_fin(const double* __restrict__ p,
                                             float* __restrict__ out) {
    __shared__ double sA[256];
    int t = threadIdx.x;
    double v[4] = {0.0, 0.0, 0.0, 0.0};
    for (int i = t; i < NB_CC; i += 256) v[0] += p[i];
    for (int i = t; i < NB_GRAD; i += 256) {
        v[1] += p[NB_CC + i];
        v[2] += p[NB_CC + NB_GRAD + i];
        v[3] += p[NB_CC + 2 * NB_GRAD + i];
    }
    double tot[4];
    for (int q = 0; q < 4; q++) {
        sA[t] = v[q];
        __syncthreads();
        for (int off = 128; off > 0; off >>= 1) {
            if (t < off) sA[t] += sA[t + off];
            __syncthreads();
        }
        tot[q] = sA[0];
        __syncthreads();
    }
    if (t == 0) {
        double ccMean = tot[0] / (double)NVOX;
        double cntD = 3.0 * (D_DIM - 1) * H_DIM * W_DIM;
        double cntH = 3.0 * D_DIM * (H_DIM - 1) * W_DIM;
        double cntW = 3.0 * D_DIM * H_DIM * (W_DIM - 1);
        double grad = (tot[1] / cntD + tot[2] / cntH + tot[3] / cntW) / 3.0;
        out[0] = (float)((1.0 - ccMean) + 0.01 * grad);
    }
}

extern "C" void kernel_launch(void* const* d_in, const int* in_sizes, int n_in,
                              void* d_out, int out_size, void* d_ws, size_t ws_size,
                              hipStream_t stream) {
    const float* I   = (const float*)d_in[0];   // y_fwd  (1,1,160,192,160)
    const float* J   = (const float*)d_in[1];   // y_inv  (1,1,160,192,160)
    const float* dsp = (const float*)d_in[2];   // dsp_fields (1,3,160,192,160)

    float*  ws  = (float*)d_ws;
    double* par = (double*)(ws + 5ull * NVOX);  // byte offset 98,304,000 (8B aligned)

    k_wh  <<<19200, 256, 0, stream>>>(I, J, ws);
    k_d_cc<<<NB_CC, 256, 0, stream>>>(ws, par);
    k_grad<<<NB_GRAD, 256, 0, stream>>>(dsp, par);
    k_fin <<<1, 256, 0, stream>>>(par, (float*)d_out);
}